// PiecewisePolynomial_19164144075551
// MI455X (gfx1250) — compile-verified
//
#include <hip/hip_runtime.h>

typedef __attribute__((ext_vector_type(2))) float v2f;
typedef __attribute__((ext_vector_type(8))) float v8f;

#define IN_F   128
#define OUT_F  256
#define BATCH  512
#define WCOLS  41     // (N-1)*SEGMENTS + 1
#define NSTEP  11     // K padded to 44 = 11 * 4
#define CROW   68     // coef row stride in floats: bank=(4m+k)%64 -> conflict-free
                      // b64 reads; 272B row stride keeps 8B alignment.

// One block = 8 waves (256 threads) per 16(b) x 16(o) output tile.
// Each wave handles 16 of the 128 input features. Per feature:
//   - basis (6 Lagrange values) computed in VALU,
//   - scattered into an LDS coefficient row (A-matrix, sparse in K),
//   - A pairs read back with conflict-free ds_load_b64 in WMMA A-layout,
//   - 11 chained V_WMMA_F32_16X16X4_F32 accumulate the 16x16 assemble tile,
//   - tile folded into running sum (+) and product (*) accumulators.
// Wave partials combine through LDS at the end.
__global__ __launch_bounds__(256)
void pp_wmma_kernel(const float* __restrict__ x,
                    const float* __restrict__ w,
                    const float* __restrict__ w_sum,
                    const float* __restrict__ w_prod,
                    float* __restrict__ out)
{
    __shared__ __align__(16) float cbuf[8][16 * CROW];  // 34,816 B
    __shared__ float lds_s[8][256];                     //  8,192 B
    __shared__ float lds_p[8][256];                     //  8,192 B

    const int tid  = threadIdx.x;
    const int wave = tid >> 5;
    const int lane = tid & 31;
    const int m    = lane & 15;   // A: M index (b row); B: N index (o col)
    const int half = lane >> 4;   // 0: lanes 0-15, 1: lanes 16-31
    const int o0   = blockIdx.x * 16;
    const int b0   = blockIdx.y * 16;

    const float* xrow = x + (size_t)(b0 + m) * IN_F;           // A-side: b row
    const float* wrow = w + (size_t)(o0 + m) * IN_F * WCOLS;   // B-side: o row

    // This lane's coefficient row in LDS (both halves share row m).
    float* crow = &cbuf[wave][m * CROW];

    // Zero the wave's coef buffer once (each half zeros 34 floats of its row;
    // intra-wave LDS ops are in-order, so no barrier is needed).
    {
        float* z = crow + half * 34;
#pragma unroll
        for (int q = 0; q < 17; ++q) {
            *(v2f*)(z + 2 * q) = (v2f){0.0f, 0.0f};
        }
    }

    float acc_s[8], acc_p[8];
#pragma unroll
    for (int r = 0; r < 8; ++r) { acc_s[r] = 0.0f; acc_p[r] = 1.0f; }

    const int i_begin = wave * 16;
#pragma unroll 1
    for (int ii = 0; ii < 16; ++ii) {
        const int i = i_begin + ii;

        // ---- segment + rescale (matches reference truncation & clamp) ----
        const float xv = xrow[i];
        int seg = (int)((xv + 1.0f) * 4.0f);
        seg = seg < 0 ? 0 : (seg > 7 ? 7 : seg);
        const float x_min = (float)seg * 0.25f - 1.0f;
        const float x_in  = (xv - x_min) * 8.0f - 1.0f;
        const int   wid   = seg * 5;                   // 0..35

        // ---- 6-point Lagrange basis on CGL nodes -cos(j*pi/5) ----
        const float dd0 = x_in + 1.0f;
        const float dd1 = x_in + 0.80901699f;
        const float dd2 = x_in + 0.30901699f;
        const float dd3 = x_in - 0.30901699f;
        const float dd4 = x_in - 0.80901699f;
        const float dd5 = x_in - 1.0f;
        const float pre2 = dd0 * dd1;
        const float pre3 = pre2 * dd2;
        const float pre4 = pre3 * dd3;
        const float pre5 = pre4 * dd4;
        const float suf3 = dd5 * dd4;
        const float suf2 = suf3 * dd3;
        const float suf1 = suf2 * dd2;
        const float suf0 = suf1 * dd1;
        // 1/denominators for CGL-6: {-1.6, 3.2, -3.2, 3.2, -3.2, 1.6}
        const float bs0 = suf0 * -1.6f;
        const float bs1 = dd0 * suf1 * 3.2f;
        const float bs2 = pre2 * suf2 * -3.2f;
        const float bs3 = pre3 * suf3 * 3.2f;
        const float bs4 = pre4 * dd5 * -3.2f;
        const float bs5 = pre5 * 1.6f;

        // ---- scatter basis into LDS coef row (halves write identical data) --
        crow[wid + 0] = bs0;
        crow[wid + 1] = bs1;
        crow[wid + 2] = bs2;
        crow[wid + 3] = bs3;
        crow[wid + 4] = bs4;
        crow[wid + 5] = bs5;

        // ---- gather A pairs (LDS, WMMA A-layout) and B pairs (global w) ----
        const float* wcol = wrow + (size_t)i * WCOLS;
        const float* arow = crow + 2 * half;           // K = 4t + 2*half (+1)
        float ca[NSTEP], cb[NSTEP], wa[NSTEP], wb[NSTEP];
#pragma unroll
        for (int t = 0; t < NSTEP; ++t) {
            v2f a = *(const v2f*)(arow + 4 * t);       // ds_load_b64, 8B aligned
            ca[t] = a.x;
            cb[t] = a.y;
            const int k0 = 4 * t + 2 * half;
            const int k1 = k0 + 1;
            // clamp padded K (41..43) to 40: coef there is 0, value is killed
            wa[t] = wcol[k0 < WCOLS ? k0 : (WCOLS - 1)];
            wb[t] = wcol[k1 < WCOLS ? k1 : (WCOLS - 1)];
        }

        // ---- restore zeros (LDS pipe is in-order: reads happen first) ----
        crow[wid + 0] = 0.0f;
        crow[wid + 1] = 0.0f;
        crow[wid + 2] = 0.0f;
        crow[wid + 3] = 0.0f;
        crow[wid + 4] = 0.0f;
        crow[wid + 5] = 0.0f;

        // ---- assemble tile = Coef[16x44] x W[44x16] via 11 chained WMMAs ----
        v8f acc = {0.f, 0.f, 0.f, 0.f, 0.f, 0.f, 0.f, 0.f};
#pragma unroll
        for (int t = 0; t < NSTEP; ++t) {
            v2f A = {ca[t], cb[t]};
            v2f B = {wa[t], wb[t]};
            acc = __builtin_amdgcn_wmma_f32_16x16x4_f32(
                /*neg_a=*/false, A, /*neg_b=*/false, B,
                /*c_mod=*/(short)0, acc, /*reuse_a=*/false, /*reuse_b=*/false);
        }

        // ---- fold this feature into running sum / product ----
#pragma unroll
        for (int r = 0; r < 8; ++r) { acc_s[r] += acc[r]; acc_p[r] *= acc[r]; }
    }

    // ---- cross-wave combine via LDS ----
#pragma unroll
    for (int r = 0; r < 8; ++r) {
        lds_s[wave][r * 32 + lane] = acc_s[r];
        lds_p[wave][r * 32 + lane] = acc_p[r];
    }
    __syncthreads();

    // 256 output elements per tile -> one per thread.
    const int r  = tid >> 5;    // C/D VGPR index
    const int l2 = tid & 31;
    float fs = 0.0f, fp = 1.0f;
#pragma unroll
    for (int wv = 0; wv < 8; ++wv) {
        fs += lds_s[wv][r * 32 + l2];
        fp *= lds_p[wv][r * 32 + l2];
    }
    // C/D layout: VGPR r, lanes 0-15 -> M=r ; lanes 16-31 -> M=r+8 ; N=lane%16
    const int o = o0 + (l2 & 15);
    const int b = b0 + r + 8 * (l2 >> 4);
    out[(size_t)b * OUT_F + o] = fs * w_sum[o] + fp * w_prod[o];
}

extern "C" void kernel_launch(void* const* d_in, const int* in_sizes, int n_in,
                              void* d_out, int out_size, void* d_ws, size_t ws_size,
                              hipStream_t stream) {
    const float* x      = (const float*)d_in[0];   // (512, 128)
    const float* w      = (const float*)d_in[1];   // (256, 128, 41)
    const float* w_sum  = (const float*)d_in[2];   // (256,)
    const float* w_prod = (const float*)d_in[3];   // (256,)
    float* out = (float*)d_out;                    // (512, 256)

    dim3 grid(OUT_F / 16, BATCH / 16);             // (16, 32) tiles
    pp_wmma_kernel<<<grid, dim3(256), 0, stream>>>(x, w, w_sum, w_prod, out);
}